// SecondaryCapsuleLayer_4793183502591
// MI455X (gfx1250) — compile-verified
//
#include <hip/hip_runtime.h>

// Problem constants (from reference setup_inputs)
#define BATCH 32
#define NL    256
#define NH    16
#define DIN   16
#define DOUT  32
#define KNUM  8
#define DCAP  512   // DIN*DOUT
#define EPSQ  1e-11f

typedef __attribute__((ext_vector_type(2))) float v2f;
typedef __attribute__((ext_vector_type(8))) float v8f;

// ---------------------------------------------------------------------------
// WMMA fragment loaders for V_WMMA_F32_16X16X4_F32 (wave32).
// A (16x4, MxK): lanes 0-15 hold M=0..15 with K={0,1}; lanes 16-31 K={2,3}.
// B (4x16, KxN): VGPR0 holds K=0 (lanes 0-15) / K=2 (lanes 16-31); VGPR1 K=1/K=3.
// C/D (16x16):   VGPR j holds row M=j (lanes 0-15) and M=j+8 (lanes 16-31).
// ---------------------------------------------------------------------------
__device__ __forceinline__ void load_a_frags(const float* __restrict__ X, int lane,
                                             float scale, v2f a[4]) {
  const int row  = lane & 15;
  const int koff = (lane >> 4) << 1;  // 0 for lanes 0-15, 2 for lanes 16-31
#pragma unroll
  for (int kk = 0; kk < 4; ++kk) {
    const int k0 = (kk << 2) + koff;
    a[kk].x = X[row * DIN + k0] * scale;
    a[kk].y = X[row * DIN + k0 + 1] * scale;
  }
}

__device__ __forceinline__ void load_b_frags(const float* __restrict__ T, int lane,
                                             int ntile, v2f bf[4]) {
  const int col  = (ntile << 4) + (lane & 15);
  const int koff = (lane >> 4) << 1;
#pragma unroll
  for (int kk = 0; kk < 4; ++kk) {
    const int k0 = (kk << 2) + koff;
    bf[kk].x = T[k0 * DOUT + col];
    bf[kk].y = T[(k0 + 1) * DOUT + col];
  }
}

__device__ __forceinline__ float wave_reduce_add(float v) {
#pragma unroll
  for (int off = 16; off > 0; off >>= 1) v += __shfl_xor(v, off, 32);
  return v;
}

// ---------------------------------------------------------------------------
// tmp2[l,h,i,o] = sum_k alpha[l,h,k] * W[k,i,o]     (Nl*Nh*Din*Dout = 2M elems)
// ---------------------------------------------------------------------------
__global__ void k_tmp2(const float* __restrict__ W, const float* __restrict__ alpha,
                       float* __restrict__ tmp2) {
  const int idx = blockIdx.x * blockDim.x + threadIdx.x;  // [0, 2097152)
  const int o  = idx & 31;
  const int i  = (idx >> 5) & 15;
  const int lh = idx >> 9;
  const float* __restrict__ al = alpha + lh * KNUM;
  float acc = 0.f;
#pragma unroll
  for (int k = 0; k < KNUM; ++k) acc += al[k] * W[(k * DIN + i) * DOUT + o];
  tmp2[idx] = acc;
}

__global__ void k_zero(float* __restrict__ p) {
  p[blockIdx.x * blockDim.x + threadIdx.x] = 0.f;
}

// ---------------------------------------------------------------------------
// c[b,l,:] = softmax(b_log[b,l,:]) over Nh       (one thread per (b,l))
// ---------------------------------------------------------------------------
__global__ void k_softmax(const float* __restrict__ blog, float* __restrict__ c) {
  const int idx = blockIdx.x * blockDim.x + threadIdx.x;  // [0, B*NL)
  const float* __restrict__ bp = blog + idx * NH;
  float m = bp[0];
#pragma unroll
  for (int h = 1; h < NH; ++h) m = fmaxf(m, bp[h]);
  float e[NH], sum = 0.f;
#pragma unroll
  for (int h = 0; h < NH; ++h) { e[h] = __expf(bp[h] - m); sum += e[h]; }
  const float inv = 1.f / sum;
  float* __restrict__ cp = c + idx * NH;
#pragma unroll
  for (int h = 0; h < NH; ++h) cp[h] = e[h] * inv;
}

// ---------------------------------------------------------------------------
// Per (b,h): s = sum_l c[b,l,h] * (X_{b,l} @ T_{l,h})  via WMMA with c folded
// into A; then squash -> v[b,h,:].  256 threads = 8 waves; each wave owns 32 l's
// and accumulates the full l-range inside its WMMA accumulators.
// The l-loop is unrolled 2x (two independent fragment sets in flight) and the
// prefetch index is clamped (branchless) so the loop body stays a clean
// load-clause -> pk_mul -> wmma pipeline.
// ---------------------------------------------------------------------------
__global__ void k_s_squash(const float* __restrict__ x, const float* __restrict__ tmp2,
                           const float* __restrict__ c, float* __restrict__ vbuf,
                           float* __restrict__ out_v, float* __restrict__ out_a,
                           int final_pass) {
  const int bh   = blockIdx.x;           // [0, BATCH*NH)
  const int b    = bh / NH;
  const int h    = bh % NH;
  const int w    = threadIdx.x >> 5;     // wave id 0..7
  const int lane = threadIdx.x & 31;

  v8f acc0 = {}, acc1 = {};
#pragma unroll 2
  for (int l = w; l < NL; l += 8) {
    // Branchless prefetch ~2 iterations ahead (clamped in-bounds).
    const int lp = (l + 16 < NL) ? (l + 16) : l;
    __builtin_prefetch(x + (size_t)(b * NL + lp) * (DIN * DIN), 0, 1);
    __builtin_prefetch(tmp2 + (size_t)(lp * NH + h) * DCAP, 0, 1);

    const float* __restrict__ X = x + (size_t)(b * NL + l) * (DIN * DIN);
    const float* __restrict__ T = tmp2 + (size_t)(l * NH + h) * DCAP;
    const float cv = c[(b * NL + l) * NH + h];

    v2f a[4], bf0[4], bf1[4];
    load_a_frags(X, lane, cv, a);
    load_b_frags(T, lane, 0, bf0);
    load_b_frags(T, lane, 1, bf1);
#pragma unroll
    for (int kk = 0; kk < 4; ++kk) {
      acc0 = __builtin_amdgcn_wmma_f32_16x16x4_f32(false, a[kk], false, bf0[kk],
                                                   (short)0, acc0, false, false);
      acc1 = __builtin_amdgcn_wmma_f32_16x16x4_f32(false, a[kk], false, bf1[kk],
                                                   (short)0, acc1, false, false);
    }
  }

  // Cross-wave reduction of the 16x32 partial tiles.
  __shared__ float sred[8 * DCAP];
  __shared__ float red[8];
  {
    const int half = lane >> 4, col = lane & 15;
#pragma unroll
    for (int j = 0; j < 8; ++j) {
      const int r = j + (half << 3);
      sred[w * DCAP + r * DOUT + col]      = acc0[j];
      sred[w * DCAP + r * DOUT + 16 + col] = acc1[j];
    }
  }
  __syncthreads();

  const int e0 = threadIdx.x, e1 = threadIdx.x + 256;
  float s0 = 0.f, s1 = 0.f;
#pragma unroll
  for (int ww = 0; ww < 8; ++ww) { s0 += sred[ww * DCAP + e0]; s1 += sred[ww * DCAP + e1]; }

  float part = wave_reduce_add(s0 * s0 + s1 * s1);
  if (lane == 0) red[w] = part;
  __syncthreads();
  float sq = 0.f;
#pragma unroll
  for (int ww = 0; ww < 8; ++ww) sq += red[ww];

  const float scale = sq / (1.f + sq);
  const float inv   = scale * rsqrtf(sq + EPSQ);   // v = s * scale / sqrt(sq+eps)
  const float v0 = s0 * inv, v1 = s1 * inv;

  float* __restrict__ vp = vbuf + (size_t)bh * DCAP;
  vp[e0] = v0; vp[e1] = v1;
  if (final_pass) {
    float* __restrict__ op = out_v + (size_t)bh * DCAP;
    op[e0] = v0; op[e1] = v1;
    if (threadIdx.x == 0) out_a[bh] = scale;
  }
}

// ---------------------------------------------------------------------------
// b_log[b,l,h] += dot(u_hat[b,l,h,:], v[b,h,:]) with u_hat recomputed via WMMA.
// One wave per (b,l); loops over the 16 higher capsules reusing the A operand.
// ---------------------------------------------------------------------------
__global__ void k_bupdate(const float* __restrict__ x, const float* __restrict__ tmp2,
                          const float* __restrict__ vbuf, float* __restrict__ blog) {
  const int w    = threadIdx.x >> 5;
  const int lane = threadIdx.x & 31;
  const int gw   = blockIdx.x * 8 + w;   // [0, BATCH*NL)
  const int b    = gw >> 8;              // / NL
  const int l    = gw & (NL - 1);

  const float* __restrict__ X = x + (size_t)(b * NL + l) * (DIN * DIN);
  v2f a[4];
  load_a_frags(X, lane, 1.f, a);

  const int half = lane >> 4, col = lane & 15;
  for (int h = 0; h < NH; ++h) {
    const float* __restrict__ T = tmp2 + (size_t)(l * NH + h) * DCAP;
    v2f bf0[4], bf1[4];
    load_b_frags(T, lane, 0, bf0);
    load_b_frags(T, lane, 1, bf1);
    v8f acc0 = {}, acc1 = {};
#pragma unroll
    for (int kk = 0; kk < 4; ++kk) {
      acc0 = __builtin_amdgcn_wmma_f32_16x16x4_f32(false, a[kk], false, bf0[kk],
                                                   (short)0, acc0, false, false);
      acc1 = __builtin_amdgcn_wmma_f32_16x16x4_f32(false, a[kk], false, bf1[kk],
                                                   (short)0, acc1, false, false);
    }
    const float* __restrict__ vv = vbuf + (size_t)(b * NH + h) * DCAP;
    float part = 0.f;
#pragma unroll
    for (int j = 0; j < 8; ++j) {
      const int r = j + (half << 3);
      part += acc0[j] * vv[r * DOUT + col];
      part += acc1[j] * vv[r * DOUT + 16 + col];
    }
    part = wave_reduce_add(part);
    if (lane == 0) blog[(size_t)(b * NL + l) * NH + h] += part;
  }
}

// ---------------------------------------------------------------------------
extern "C" void kernel_launch(void* const* d_in, const int* in_sizes, int n_in,
                              void* d_out, int out_size, void* d_ws, size_t ws_size,
                              hipStream_t stream) {
  const float* x     = (const float*)d_in[0];   // (B, Nl, Din, Din)
  // d_in[1] = adj, unused by forward
  const float* W     = (const float*)d_in[2];   // (K, Din, Dout)
  const float* alpha = (const float*)d_in[3];   // (Nl, Nh, K)

  float* out   = (float*)d_out;
  float* out_v = out;                             // B*Nh*D = 262144 floats
  float* out_a = out + (size_t)BATCH * NH * DCAP; // 512 floats

  float* ws   = (float*)d_ws;                     // total ~10.5 MB of f32 scratch
  float* tmp2 = ws;                               // Nl*Nh*Din*Dout = 2,097,152
  float* blog = tmp2 + (size_t)NL * NH * DIN * DOUT;  // B*Nl*Nh = 131,072
  float* cbuf = blog + (size_t)BATCH * NL * NH;       // 131,072
  float* vbuf = cbuf + (size_t)BATCH * NL * NH;       // B*Nh*D = 262,144

  // 1) tmp2 = einsum('kio,lhk->lhio')
  k_tmp2<<<(NL * NH * DIN * DOUT) / 256, 256, 0, stream>>>(W, alpha, tmp2);
  // 2) routing logits b = 0
  k_zero<<<(BATCH * NL * NH) / 256, 256, 0, stream>>>(blog);

  // 3) dynamic routing: 3 coupling passes, 2 agreement updates
  for (int it = 0; it < 3; ++it) {
    k_softmax<<<(BATCH * NL) / 256, 256, 0, stream>>>(blog, cbuf);
    const int final_pass = (it == 2);
    k_s_squash<<<BATCH * NH, 256, 0, stream>>>(x, tmp2, cbuf, vbuf, out_v, out_a,
                                               final_pass);
    if (!final_pass)
      k_bupdate<<<(BATCH * NL) / 8, 256, 0, stream>>>(x, tmp2, vbuf, blog);
  }
}